// InverseWaveletPatcher_65420941852885
// MI455X (gfx1250) — compile-verified
//
#include <hip/hip_runtime.h>
#include <stdint.h>

// ---------------- problem geometry ----------------
#define T_LEN       4096      // samples per signal (power of two -> cheap wrap)
#define W_TILE      128       // output samples produced per workgroup
#define N_ROWS      16        // signals per workgroup (WMMA M dimension)
#define LDS_STRIDE  189       // max halo width 188, +1 -> odd stride => conflict-free A reads

typedef __attribute__((ext_vector_type(2))) float v2f;
typedef __attribute__((ext_vector_type(8))) float v8f;

// db4 synthesis filters; rec_hi[k] = (-1)^k * rec_lo[7-k]
constexpr float REC_LO[8] = {
    0.23037781330885523f,  0.7148465705525415f,  0.6308807679295904f,
   -0.02798376941698385f, -0.18703481171888114f, 0.030841381835986965f,
    0.032883011666982945f, -0.010597401784997278f };
constexpr float REC_HI[8] = {
   -0.010597401784997278f, -0.032883011666982945f, 0.030841381835986965f,
    0.18703481171888114f,  -0.02798376941698385f, -0.6308807679295904f,
    0.7148465705525415f,   -0.23037781330885523f };

// Low 32 bits of a flat pointer into LDS are the LDS byte offset
// (aperture lives in ADDR[63:32], per CDNA5 aperture rules).
__device__ __forceinline__ unsigned lds_byte_off(const float* p) {
    return (unsigned)(uintptr_t)p;
}

// Async DMA one coefficient plane tile (16 rows x width, wrapped mod T) into LDS.
// Tracked with ASYNCcnt; caller syncs with s_wait_asynccnt + barrier.
__device__ __forceinline__ void async_plane_to_lds(
    unsigned lds_buf, const float* __restrict__ x,
    int sbase, int plane, int g0, int origin, int width, int tid)
{
#pragma unroll
    for (int r = 0; r < N_ROWS; ++r) {
        const float* src = x + ((size_t)((sbase + r) * 4 + plane)) * T_LEN;
        for (int c = tid; c < width; c += 256) {
            unsigned goff  = (unsigned)(((g0 + origin + c) & (T_LEN - 1)) * 4);
            unsigned laddr = lds_buf + (unsigned)((r * LDS_STRIDE + c) * 4);
            asm volatile("global_load_async_to_lds_b32 %0, %1, %2"
                         :: "v"(laddr), "v"(goff), "s"(src) : "memory");
        }
    }
}

// One ISWT level as banded WMMA:  out(16x16) = sum_b A_b(16x4) * B_b(4x16).
//   A fragment: lane l -> M=row=l&15, K = ((l>>4)*2)+h  (ISA 16x4 f32 A layout)
//   B fragment: lane l -> N=j=l&15,  K = ((l>>4)*2)+h   (C/D-analogous striping)
//   D fragment: VGPR v -> M = v + (l>=16 ? 8:0), N = l&15
// Input buffers have origin = outOrigin - 4D, so colIn = 16*ti + 4*b + K.
template<int D, int NBLK, int NTILES, bool TO_GLOBAL>
__device__ __forceinline__ void level_compute(
    const float* __restrict__ sIn, const float* __restrict__ sCd,
    float* __restrict__ sOut, float* __restrict__ gOut, int g0)
{
    const int lane = threadIdx.x & 31;
    const int wave = threadIdx.x >> 5;
    const int j    = lane & 15;
    const int ksel = (lane >> 4) << 1;
    constexpr int KMIN = -4 * D;

    // Per-level banded filter fragments (compile-time folded selects -> registers)
    v2f blo[NBLK], bhi[NBLK];
#pragma unroll
    for (int b = 0; b < NBLK; ++b) {
#pragma unroll
        for (int h = 0; h < 2; ++h) {
            const int kg = KMIN + 4 * b + ksel + h;
            float vlo = 0.f, vhi = 0.f;
#pragma unroll
            for (int m = 0; m < 8; ++m) {
                if (j - D * (m - 3) == kg) { vlo += REC_LO[m]; vhi += REC_HI[m]; }
            }
            blo[b][h] = vlo; bhi[b][h] = vhi;
        }
    }

    const int row = lane & 15;
    for (int ti = wave; ti < NTILES; ti += 8) {      // wave-uniform => EXEC all-ones
        v8f acc = {0.f,0.f,0.f,0.f,0.f,0.f,0.f,0.f};
#pragma unroll
        for (int b = 0; b < NBLK; ++b) {
            const int col = ti * 16 + 4 * b + ksel;
            const float* pa = sIn + row * LDS_STRIDE + col;
            const float* pc = sCd + row * LDS_STRIDE + col;
            v2f a_res = { pa[0], pa[1] };
            v2f a_cd  = { pc[0], pc[1] };
            acc = __builtin_amdgcn_wmma_f32_16x16x4_f32(false, a_res, false, blo[b],
                                                        (short)0, acc, false, false);
            acc = __builtin_amdgcn_wmma_f32_16x16x4_f32(false, a_cd,  false, bhi[b],
                                                        (short)0, acc, false, false);
        }
        const int mb = (lane >> 4) << 3;
        if (TO_GLOBAL) {
#pragma unroll
            for (int v = 0; v < 8; ++v)
                gOut[(size_t)(mb + v) * T_LEN + (g0 + ti * 16 + j)] = 0.5f * acc[v];
        } else {
#pragma unroll
            for (int v = 0; v < 8; ++v)
                sOut[(mb + v) * LDS_STRIDE + ti * 16 + j] = 0.5f * acc[v];
        }
    }
}

// grid = (4096/128 time tiles, 2048/16 row groups); block = 256 threads (8 waves)
__global__ __launch_bounds__(256) void iswt_db4_l3_kernel(
    const float* __restrict__ x, float* __restrict__ out)
{
    __shared__ float smem[4 * N_ROWS * LDS_STRIDE];   // 48.4 KB
    float* P  = smem + 0 * N_ROWS * LDS_STRIDE;       // cA, later level-2 output
    float* R  = smem + 1 * N_ROWS * LDS_STRIDE;       // level-1 output
    float* C0 = smem + 2 * N_ROWS * LDS_STRIDE;       // cD3, later cD1
    float* C1 = smem + 3 * N_ROWS * LDS_STRIDE;       // cD2

    const int tid   = threadIdx.x;
    const int g0    = blockIdx.x * W_TILE;            // output start in time
    const int sbase = blockIdx.y * N_ROWS;            // first signal of this group

    // Stage cA, cD3, cD2 (halo widths 188/188/160; origins -28/-28/-12)
    async_plane_to_lds(lds_byte_off(P),  x, sbase, 0, g0, -28, 188, tid);
    async_plane_to_lds(lds_byte_off(C0), x, sbase, 1, g0, -28, 188, tid);
    async_plane_to_lds(lds_byte_off(C1), x, sbase, 2, g0, -12, 160, tid);
    asm volatile("s_wait_asynccnt 0" ::: "memory");
    __syncthreads();

    // Level 1: d=4, 10 tiles of 16 (out origin -12), P,C0 -> R
    level_compute<4, 11, 10, false>(P, C0, R, nullptr, g0);
    __syncthreads();

    // C0 is now free: prefetch cD1 (origin -4, width 136) under level-2 compute
    async_plane_to_lds(lds_byte_off(C0), x, sbase, 3, g0, -4, 136, tid);

    // Level 2: d=2, 9 tiles (out origin -4), R,C1 -> P
    level_compute<2, 8, 9, false>(R, C1, P, nullptr, g0);
    asm volatile("s_wait_asynccnt 0" ::: "memory");
    __syncthreads();

    // Level 3: d=1, 8 tiles (out origin 0), P,C0 -> global
    level_compute<1, 6, 8, true>(P, C0, nullptr, out + (size_t)sbase * T_LEN, g0);
}

extern "C" void kernel_launch(void* const* d_in, const int* in_sizes, int n_in,
                              void* d_out, int out_size, void* d_ws, size_t ws_size,
                              hipStream_t stream) {
    const float* x  = (const float*)d_in[0];   // [32, 256, 4096] f32
    float* out      = (float*)d_out;           // [32, 64, 4096] f32
    dim3 grid(T_LEN / W_TILE, 2048 / N_ROWS);  // (32, 128)
    iswt_db4_l3_kernel<<<grid, 256, 0, stream>>>(x, out);
    (void)in_sizes; (void)n_in; (void)out_size; (void)d_ws; (void)ws_size;
}